// MirrorPDHG_18313740550348
// MI455X (gfx1250) — compile-verified
//
#include <hip/hip_runtime.h>

// WMF entropic mirror-prox step for MI455X (gfx1250, wave32).
// One workgroup (256 threads / 8 waves) per token n.
// T = M[Kset[n]] staged once in LDS via async global->LDS copies (ASYNCcnt),
// Gram G = T*T^T via V_WMMA_F32_16X16X4_F32 with 4 independent accumulators.

typedef __attribute__((ext_vector_type(2))) float v2f;
typedef __attribute__((ext_vector_type(8))) float v8f;
typedef __attribute__((ext_vector_type(4))) int   v4i;

typedef __attribute__((address_space(1))) v4i* gv4i_p;   // global 16B chunk
typedef __attribute__((address_space(3))) v4i* lv4i_p;   // LDS 16B chunk

#define K_NEI 64
#define DDIM  512
#define TSTR  516          // padded LDS row stride (floats): 516 % 64 == 4 -> conflict-free WMMA operand reads
#define BETA  1.0f
#define TAU   0.05f
#define RHO   0.1f
#define EPSL  1e-12f

#if __has_builtin(__builtin_amdgcn_global_load_async_to_lds_b128) && \
    __has_builtin(__builtin_amdgcn_s_wait_asynccnt)
#define USE_ASYNC_LDS 1
#else
#define USE_ASYNC_LDS 0
#endif

__global__ __launch_bounds__(256)
void MirrorPDHG_wmma_kernel(const float* __restrict__ M,
                            const float* __restrict__ P,
                            const float* __restrict__ Y,
                            const float* __restrict__ Lam,
                            const int*   __restrict__ Kset,
                            float*       __restrict__ LamNew)
{
    extern __shared__ float smem[];
    float* T   = smem;               // 64*516 floats  (~129 KB)
    float* Xi  = T  + K_NEI * TSTR;  // 512
    float* Pl  = Xi + DDIM;          // 64   P[n]
    float* sq  = Pl + K_NEI;         // 64   row squared norms
    float* sc  = sq + K_NEI;         // 64   scores
    float* sm  = sc + K_NEI;         // 64   smooth = cost @ P
    float* lg  = sm + K_NEI;         // 64   logits -> P_new
    float* red = lg + K_NEI;         // 64   reduction scratch

    const int n   = blockIdx.x;
    const int tid = threadIdx.x;

    const float* Yr = Y    + (size_t)n * DDIM;
    const float* Lr = Lam  + (size_t)n * DDIM;
    const int*   Kr = Kset + (size_t)n * K_NEI;
    float*       Or = LamNew + (size_t)n * DDIM;

    // ---- init accumulators + load P[n] ----
    if (tid < K_NEI) {
        Pl[tid] = P[(size_t)n * K_NEI + tid];
        sq[tid] = 0.f; sc[tid] = 0.f; sm[tid] = 0.f;
    }

    // ---- gather T = M[Kset[n]] into LDS: 4 threads per row, 16B chunks ----
    {
        const int row  = tid >> 2;
        const int part = tid & 3;
        const int gidx = Kr[row];
        const float* __restrict__ src = M + (size_t)gidx * DDIM;
        float* dstrow = T + row * TSTR;              // row*2064 B: 16B aligned
#if USE_ASYNC_LDS
        #pragma unroll
        for (int i = 0; i < 32; ++i) {
            const int q = part + 4 * i;              // float4 index within row
            __builtin_amdgcn_global_load_async_to_lds_b128(
                (gv4i_p)(src + 4 * q),
                (lv4i_p)(dstrow + 4 * q),
                0, 0);
        }
        __builtin_amdgcn_s_wait_asynccnt(0);
#else
        const float4* __restrict__ s4 = (const float4*)src;
        float4* d4 = (float4*)dstrow;
        #pragma unroll
        for (int i = 0; i < 32; ++i) {
            const int q = part + 4 * i;
            d4[q] = s4[q];
        }
#endif
    }
    __syncthreads();

    // ---- phase 2: Y_from_P = P.T (per column), Xi = Lam + rho*(Y - Y_from_P) ----
    const int d0 = tid, d1 = tid + 256;
    const float y0 = Yr[d0], y1 = Yr[d1];
    const float l0 = Lr[d0], l1 = Lr[d1];
    {
        float a0 = 0.f, a1 = 0.f;
        #pragma unroll 8
        for (int kk = 0; kk < K_NEI; ++kk) {
            const float p = Pl[kk];
            a0 += p * T[kk * TSTR + d0];
            a1 += p * T[kk * TSTR + d1];
        }
        Xi[d0] = l0 + RHO * (y0 - a0);
        Xi[d1] = l1 + RHO * (y1 - a1);
    }
    __syncthreads();

    // ---- phase 3: sq[i] = |T_i|^2 and scores[i] = T_i . Xi (fused) ----
    {
        const int row = tid >> 2, part = tid & 3;
        const float* Trow = T + row * TSTR;
        float s2 = 0.f, sx = 0.f;
        #pragma unroll 8
        for (int m = 0; m < 128; ++m) {
            const int d = part + 4 * m;           // strided: conflict-free banks
            const float v = Trow[d];
            s2 += v * v;
            sx += v * Xi[d];
        }
        atomicAdd(&sq[row], s2);
        atomicAdd(&sc[row], sx);
    }
    __syncthreads();

    // ---- phase 4: G = T*T^T via V_WMMA_F32_16X16X4_F32 (4 accumulators to
    //      break the D->C dependency chain); fold cost into smooth ----
    {
        const int wave = tid >> 5;
        const int lane = tid & 31;
        const int L16  = lane & 15, Lh = lane >> 4;
        #pragma unroll
        for (int half = 0; half < 2; ++half) {
            const int tt = wave + 8 * half;           // 16 tiles, 2 per wave
            const int i0 = (tt >> 2) * 16;
            const int j0 = (tt & 3) * 16;
            // f32 WMMA A(16x4) layout == B(4x16) layout for B = A^T (Gram):
            // lane holds T[blk + L16][k + 2*Lh], T[blk + L16][k + 2*Lh + 1]
            const float* Arow = T + (i0 + L16) * TSTR + 2 * Lh;
            const float* Brow = T + (j0 + L16) * TSTR + 2 * Lh;
            v8f acc0 = {}, acc1 = {}, acc2 = {}, acc3 = {};
            #pragma unroll 4
            for (int kk = 0; kk < DDIM; kk += 16) {
                const v2f a0 = *(const v2f*)(Arow + kk);
                const v2f b0 = *(const v2f*)(Brow + kk);
                const v2f a1 = *(const v2f*)(Arow + kk + 4);
                const v2f b1 = *(const v2f*)(Brow + kk + 4);
                const v2f a2 = *(const v2f*)(Arow + kk + 8);
                const v2f b2 = *(const v2f*)(Brow + kk + 8);
                const v2f a3 = *(const v2f*)(Arow + kk + 12);
                const v2f b3 = *(const v2f*)(Brow + kk + 12);
                acc0 = __builtin_amdgcn_wmma_f32_16x16x4_f32(
                         false, a0, false, b0, (short)0, acc0, false, false);
                acc1 = __builtin_amdgcn_wmma_f32_16x16x4_f32(
                         false, a1, false, b1, (short)0, acc1, false, false);
                acc2 = __builtin_amdgcn_wmma_f32_16x16x4_f32(
                         false, a2, false, b2, (short)0, acc2, false, false);
                acc3 = __builtin_amdgcn_wmma_f32_16x16x4_f32(
                         false, a3, false, b3, (short)0, acc3, false, false);
            }
            // C/D layout: acc[r] = G[i0 + r + 8*Lh][j0 + L16]
            const int   j   = j0 + L16;
            const float pj  = Pl[j];
            const float sqj = sq[j];
            #pragma unroll
            for (int r = 0; r < 8; ++r) {
                const int i = i0 + r + 8 * Lh;
                const float g = (acc0[r] + acc1[r]) + (acc2[r] + acc3[r]);
                float cost = sq[i] + sqj - 2.0f * g;
                cost = fmaxf(cost, 0.0f);
                atomicAdd(&sm[i], cost * pj);
            }
        }
    }
    __syncthreads();

    // ---- phase 5: logits + 64-wide softmax ----
    if (tid < K_NEI) {
        const float lgt = __logf(Pl[tid] + EPSL)
                        - (BETA * sc[tid] + TAU * sm[tid]) / BETA;
        lg[tid]  = lgt;
        red[tid] = lgt;
    }
    __syncthreads();
    for (int s = 32; s > 0; s >>= 1) {
        if (tid < s) red[tid] = fmaxf(red[tid], red[tid + s]);
        __syncthreads();
    }
    const float mx = red[0];
    __syncthreads();
    if (tid < K_NEI) {
        const float e = __expf(lg[tid] - mx);
        lg[tid]  = e;
        red[tid] = e;
    }
    __syncthreads();
    for (int s = 32; s > 0; s >>= 1) {
        if (tid < s) red[tid] += red[tid + s];
        __syncthreads();
    }
    const float inv = 1.0f / red[0];
    __syncthreads();
    if (tid < K_NEI) lg[tid] *= inv;          // lg now holds P_new
    __syncthreads();

    // ---- phase 6: Lam_new = Lam + rho*(Y - P_new . T) ----
    {
        float a0 = 0.f, a1 = 0.f;
        #pragma unroll 8
        for (int kk = 0; kk < K_NEI; ++kk) {
            const float p = lg[kk];
            a0 += p * T[kk * TSTR + d0];
            a1 += p * T[kk * TSTR + d1];
        }
        Or[d0] = l0 + RHO * (y0 - a0);
        Or[d1] = l1 + RHO * (y1 - a1);
    }
}

extern "C" void kernel_launch(void* const* d_in, const int* in_sizes, int n_in,
                              void* d_out, int out_size, void* d_ws, size_t ws_size,
                              hipStream_t stream) {
    (void)n_in; (void)out_size; (void)d_ws; (void)ws_size;
    const float* M    = (const float*)d_in[0];
    const float* P    = (const float*)d_in[1];
    const float* Y    = (const float*)d_in[2];
    const float* Lam  = (const float*)d_in[3];
    const int*   Kset = (const int*)d_in[4];
    float*       out  = (float*)d_out;

    const int n_tok = in_sizes[1] / K_NEI;    // 4096
    const size_t shmem = (size_t)(K_NEI * TSTR + DDIM + 6 * K_NEI) * sizeof(float);

    // > 64KB dynamic LDS: raise the per-kernel cap (deterministic, idempotent)
    (void)hipFuncSetAttribute(reinterpret_cast<const void*>(MirrorPDHG_wmma_kernel),
                              hipFuncAttributeMaxDynamicSharedMemorySize, (int)shmem);

    MirrorPDHG_wmma_kernel<<<n_tok, 256, shmem, stream>>>(M, P, Y, Lam, Kset, out);
}